// LaneFollow_27496380629539
// MI455X (gfx1250) — compile-verified
//
#include <hip/hip_runtime.h>

typedef float v2f __attribute__((ext_vector_type(2)));
typedef float v4f __attribute__((ext_vector_type(4)));
typedef float v8f __attribute__((ext_vector_type(8)));
typedef int   v4i __attribute__((ext_vector_type(4)));

#define P_  12
#define B_  32
#define V_  32
#define F_  256
#define S_  32
#define NL_ 16
#define LF_ 64
#define T_  20
#define H_  8
#define D_  32
#define L_  512   // NL_*S_

#if __has_builtin(__builtin_amdgcn_global_load_async_to_lds_b128) && \
    __has_builtin(__builtin_amdgcn_s_wait_asynccnt)
#define HAVE_ASYNC_LDS 1
typedef __attribute__((address_space(1))) v4i glb_v4i_t;   // global (device) addr space
typedef __attribute__((address_space(3))) v4i lds_v4i_t;   // LDS addr space
#define ASYNC_CP128(gsrc, ldst) \
  __builtin_amdgcn_global_load_async_to_lds_b128((glb_v4i_t*)(gsrc), (lds_v4i_t*)(ldst), 0, 0)
#define ASYNC_WAIT() __builtin_amdgcn_s_wait_asynccnt(0)
#else
#define HAVE_ASYNC_LDS 0
#define ASYNC_CP128(gsrc, ldst) (*(v4f*)(ldst) = *(const v4f*)(gsrc))
#define ASYNC_WAIT() ((void)0)
#endif

static __device__ __forceinline__ v8f wmma_f32(v2f a, v2f b, v8f c) {
  // V_WMMA_F32_16X16X4_F32: (neg_a, A, neg_b, B, c_mod, C, reuse_a, reuse_b)
  return __builtin_amdgcn_wmma_f32_16x16x4_f32(false, a, false, b, (short)0, c, false, false);
}

// ---------------------------------------------------------------------------
// Kernel 1: KT = (lanes_f @ Wk + bk)^T per head -> (B,H,d,L)
//           LV = log_sigmoid(lanes_f @ Wv + bv) -> (B,H,L,d)
// lanes_f[b, nl*S+s, :] = lanes[s, b, nl, :]  (lane-major flatten)
// All 8 waves share one M-tile; its 16x64 A-slab is async-staged to LDS once.
// ---------------------------------------------------------------------------
__global__ void __launch_bounds__(256) proj_kv_kernel(
    const float* __restrict__ lanes,
    const float* __restrict__ Wk, const float* __restrict__ bk,
    const float* __restrict__ Wv, const float* __restrict__ bv,
    float* __restrict__ KTbuf, float* __restrict__ LVbuf) {
  __shared__ float sA[16 * LF_];            // 4 KB A-tile
  const int tid  = threadIdx.x;
  const int wave = tid >> 5;
  const int lane = tid & 31;
  const int half = lane >> 4;
  const int lm   = lane & 15;
  const int mt   = blockIdx.x >> 1;                 // 1024 M-tiles over B*L rows
  const int nt   = (blockIdx.x & 1) * 8 + wave;     // 16 N-tiles over F cols

  // Stage A-tile: 16 gathered rows x 64 contiguous floats, one b128 per thread
  {
    const int srow = tid >> 4;              // 0..15
    const int scol = (tid & 15) * 4;        // 0,4,...,60
    const int gr = mt * 16 + srow;
    const int bb = gr >> 9;                 // / L_
    const int l  = gr & (L_ - 1);
    const int nl = l >> 5;                  // / S_
    const int s  = l & (S_ - 1);
    const long g = (((long)s * B_ + bb) * NL_ + nl) * LF_ + scol;
    ASYNC_CP128(lanes + g, sA + srow * LF_ + scol);
  }
  ASYNC_WAIT();
  __syncthreads();

  const int ncol = nt * 16 + lm;
  v8f accK = {}; v8f accV = {};
#pragma unroll
  for (int kk = 0; kk < 16; ++kk) {         // LF=64, K-step 4
    const int kb = kk * 4 + half * 2;
    const v2f a = *(const v2f*)(sA + lm * LF_ + kb);              // ds_load_b64
    v2f bkf; bkf.x = Wk[kb * F_ + ncol];       bkf.y = Wk[(kb + 1) * F_ + ncol];
    v2f bvf; bvf.x = Wv[kb * F_ + ncol];       bvf.y = Wv[(kb + 1) * F_ + ncol];
    accK = wmma_f32(a, bkf, accK);
    accV = wmma_f32(a, bvf, accV);
  }
  const float biasK = bk[ncol];
  const float biasV = bv[ncol];
  const int h  = ncol >> 5;
  const int dd = ncol & 31;
#pragma unroll
  for (int rr = 0; rr < 8; ++rr) {
    const int row = mt * 16 + rr + half * 8;
    const int ob  = row >> 9;
    const int ol  = row & (L_ - 1);
    // K transposed: (B,H,d,L) so attention B-fragments are lane-contiguous
    KTbuf[(((long)ob * H_ + h) * D_ + dd) * L_ + ol] = accK[rr] + biasK;
    const float x = accV[rr] + biasV;
    // numerically stable log-sigmoid
    LVbuf[(((long)ob * H_ + h) * L_ + ol) * D_ + dd] =
        fminf(x, 0.0f) - log1pf(__expf(-fabsf(x)));
  }
}

// ---------------------------------------------------------------------------
// Kernel 2: per (p,b,h): scores = q@K^T, mask, softmax, exp(attn @ LV / Z)
// ---------------------------------------------------------------------------
__global__ void __launch_bounds__(256) attn_kernel(
    const float* __restrict__ query,
    const unsigned char* __restrict__ mask_input,   // (T,B,V) bool
    const unsigned char* __restrict__ mask_lanes,   // (S,B,NL) bool
    const float* __restrict__ KT, const float* __restrict__ LVbuf,
    float* __restrict__ ATT) {
  extern __shared__ float smem[];
  float* sQ    = smem;              // 32 x 32 query slice
  float* sS    = sQ + V_ * D_;      // 32 x 512 scores / exp-scores
  float* sML   = sS + 32 * L_;      // 512 column mask
  float* sMI   = sML + L_;          // 32 row mask
  float* sRed  = sMI + 32;          // 256 reduction scratch
  float* sMax  = sRed + 256;        // 32
  float* sInv  = sMax + 32;         // 32
  float* sPart = sInv + 32;         // 8 x 256 split-K partials

  const int tid  = threadIdx.x;
  const int wave = tid >> 5;
  const int lane = tid & 31;
  const int half = lane >> 4;
  const int lm   = lane & 15;

  const int p   = blockIdx.x / (B_ * H_);
  const int rem = blockIdx.x % (B_ * H_);
  const int b   = rem / H_;
  const int h   = rem % H_;

  const long qbase  = (((long)p * B_ + b) * V_) * F_ + h * D_;
  const long kvbase = ((long)b * H_ + h) * (long)L_ * D_;   // same extent for KT and LV

  // warm L2 for this block's KT/LV slabs (global_prefetch_b8)
  __builtin_prefetch(KT    + kvbase + (long)tid * 64, 0, 0);
  __builtin_prefetch(LVbuf + kvbase + (long)tid * 64, 0, 0);

  // Phase 0a: stage q slice (32 rows x 32 cols fp32) into LDS (async).
  {
    const int row = tid >> 3;           // 0..31
    const int seg = tid & 7;            // 0..7, 4 floats each
    ASYNC_CP128(query + qbase + (long)row * F_ + seg * 4, sQ + row * D_ + seg * 4);
  }

  // Phase 0b: masks. NOTE: reference flattens mask_lanes POSITION-major
  // (j = s*NL + nl) while K/V use lane-major l = nl*S + s; replicate that.
  for (int j = tid; j < L_; j += 256) {
    const int sm  = j >> 4;   // j / NL
    const int nlm = j & 15;   // j % NL
    sML[j] = mask_lanes[((long)sm * B_ + b) * NL_ + nlm] ? 1.0f : 0.0f;
  }
  if (tid < V_) {
    int any = 0;
    for (int t = 0; t < T_; ++t) any |= mask_input[((long)t * B_ + b) * V_ + tid];
    sMI[tid] = any ? 1.0f : 0.0f;
  }
  ASYNC_WAIT();
  __syncthreads();

  // Phase A: scores (32 x 512) via WMMA. A from LDS, B lane-contiguous from KT.
#pragma unroll
  for (int t8 = 0; t8 < 8; ++t8) {
    const int t  = wave + t8 * 8;       // 64 tiles / 8 waves
    const int vt = t >> 5;
    const int lt = t & 31;
    const int col = lt * 16 + lm;
    v8f acc = {};
#pragma unroll
    for (int kk = 0; kk < 8; ++kk) {    // d=32, K-step 4
      const int kb = kk * 4 + half * 2;
      const v2f a = *(const v2f*)(sQ + (vt * 16 + lm) * D_ + kb);   // ds_load_b64
      v2f bf; bf.x = KT[kvbase + (long)kb * L_ + col];              // coalesced
              bf.y = KT[kvbase + (long)(kb + 1) * L_ + col];
      acc = wmma_f32(a, bf, acc);
    }
    const float mlc = sML[col];
#pragma unroll
    for (int rr = 0; rr < 8; ++rr) {
      const int row = vt * 16 + rr + half * 8;
      sS[row * L_ + col] = (mlc * sMI[row] != 0.0f) ? acc[rr] : -1e9f;
    }
  }
  __syncthreads();

  // Phase B: softmax over L (store unnormalized exp, keep 1/Z per row)
  const int rrow = tid >> 3;
  const int g    = tid & 7;
  float pmax = -3.4e38f;
  for (int c = g; c < L_; c += 8) pmax = fmaxf(pmax, sS[rrow * L_ + c]);
  sRed[tid] = pmax;
  __syncthreads();
  if (g == 0) {
    float m = sRed[rrow * 8];
#pragma unroll
    for (int i = 1; i < 8; ++i) m = fmaxf(m, sRed[rrow * 8 + i]);
    sMax[rrow] = m;
  }
  __syncthreads();
  const float rm = sMax[rrow];
  float psum = 0.0f;
  for (int c = g; c < L_; c += 8) {
    const float e = __expf(sS[rrow * L_ + c] - rm);
    sS[rrow * L_ + c] = e;
    psum += e;
  }
  sRed[tid] = psum;
  __syncthreads();
  if (g == 0) {
    float z = 0.0f;
#pragma unroll
    for (int i = 0; i < 8; ++i) z += sRed[rrow * 8 + i];
    sInv[rrow] = 1.0f / z;
  }
  __syncthreads();

  // Phase C: out = exp( (E @ LV) / Z ), split-K across all 8 waves:
  // tile = wave&3 (vt,nt), k-half = wave>>2; partials combined through LDS.
  {
    const int tl  = wave & 3;
    const int vt  = tl >> 1;
    const int nt  = tl & 1;
    const int kc  = wave >> 2;          // 0 or 1: K in [kc*256, kc*256+256)
    v8f acc = {};
    for (int kk = 0; kk < 64; ++kk) {
      const int kb = kc * 256 + kk * 4 + half * 2;
      const v2f a = *(const v2f*)(sS + (vt * 16 + lm) * L_ + kb);   // ds_load_b64
      v2f bf; bf.x = LVbuf[kvbase + (long)kb * D_ + nt * 16 + lm];
              bf.y = LVbuf[kvbase + (long)(kb + 1) * D_ + nt * 16 + lm];
      acc = wmma_f32(a, bf, acc);
    }
#pragma unroll
    for (int rr = 0; rr < 8; ++rr) sPart[wave * 256 + rr * 32 + lane] = acc[rr];
  }
  __syncthreads();
  if (wave < 4) {
    const int vt = wave >> 1;
    const int nt = wave & 1;
#pragma unroll
    for (int rr = 0; rr < 8; ++rr) {
      const int row = vt * 16 + rr + half * 8;
      const float v = sPart[wave * 256 + rr * 32 + lane] +
                      sPart[(wave + 4) * 256 + rr * 32 + lane];
      ATT[(((long)p * B_ + b) * V_ + row) * F_ + h * D_ + nt * 16 + lm] =
          __expf(v * sInv[row]);
    }
  }
}

// ---------------------------------------------------------------------------
// Kernel 3: out = ATT @ Wc + bc   (12288 x 256 x 256)
// 8 waves share one M-tile; its 16x256 A-slab is async-staged to LDS once.
// ---------------------------------------------------------------------------
__global__ void __launch_bounds__(256) combine_kernel(
    const float* __restrict__ ATT,
    const float* __restrict__ Wc, const float* __restrict__ bc,
    float* __restrict__ out) {
  __shared__ float sA[16 * F_];             // 16 KB A-tile
  const int tid  = threadIdx.x;
  const int wave = tid >> 5;
  const int lane = tid & 31;
  const int half = lane >> 4;
  const int lm   = lane & 15;
  const int mt   = blockIdx.x >> 1;                 // 768 M-tiles
  const int nt   = (blockIdx.x & 1) * 8 + wave;     // 16 N-tiles

  // Stage A-tile: 16 rows x 256 floats = 1024 b128 segments, 4 per thread
#pragma unroll
  for (int i = 0; i < 4; ++i) {
    const int si  = tid + 256 * i;          // 0..1023
    const int row = si >> 6;                // 0..15
    const int c4  = (si & 63) * 4;          // 0..252
    ASYNC_CP128(ATT + (long)(mt * 16 + row) * F_ + c4, sA + row * F_ + c4);
  }
  ASYNC_WAIT();
  __syncthreads();

  const int ncol = nt * 16 + lm;
  v8f acc = {};
  for (int kk = 0; kk < 64; ++kk) {         // F=256, K-step 4
    const int kb = kk * 4 + half * 2;
    const v2f a = *(const v2f*)(sA + lm * F_ + kb);               // ds_load_b64
    v2f bf; bf.x = Wc[kb * F_ + ncol];       bf.y = Wc[(kb + 1) * F_ + ncol];
    acc = wmma_f32(a, bf, acc);
  }
  const float bias = bc[ncol];
#pragma unroll
  for (int rr = 0; rr < 8; ++rr) {
    const long row = mt * 16 + rr + half * 8;
    out[row * F_ + ncol] = acc[rr] + bias;
  }
}

extern "C" void kernel_launch(void* const* d_in, const int* in_sizes, int n_in,
                              void* d_out, int out_size, void* d_ws, size_t ws_size,
                              hipStream_t stream) {
  const float* query = (const float*)d_in[0];
  const float* lanes = (const float*)d_in[1];
  const unsigned char* mask_input = (const unsigned char*)d_in[2];
  const unsigned char* mask_lanes = (const unsigned char*)d_in[3];
  const float* Wk = (const float*)d_in[4];
  const float* bk = (const float*)d_in[5];
  const float* Wv = (const float*)d_in[6];
  const float* bv = (const float*)d_in[7];
  const float* Wc = (const float*)d_in[8];
  const float* bc = (const float*)d_in[9];
  float* out = (float*)d_out;

  float* KTbuf = (float*)d_ws;                                   // B*H*d*L = 4M floats
  float* LVbuf = KTbuf + (size_t)B_ * H_ * L_ * D_;              // 4M floats
  float* ATT   = LVbuf + (size_t)B_ * H_ * L_ * D_;              // P*B*V*F = 3M floats

  proj_kv_kernel<<<2048, 256, 0, stream>>>(lanes, Wk, bk, Wv, bv, KTbuf, LVbuf);

  const size_t smem =
      (size_t)(V_ * D_ + 32 * L_ + L_ + 32 + 256 + 32 + 32 + 8 * 256) * sizeof(float); // ~81 KB
  attn_kernel<<<P_ * B_ * H_, 256, smem, stream>>>(query, mask_input, mask_lanes,
                                                   KTbuf, LVbuf, ATT);

  combine_kernel<<<1536, 256, 0, stream>>>(ATT, Wc, bc, out);
}